// Node_prompt_65335042506952
// MI455X (gfx1250) — compile-verified
//
#include <hip/hip_runtime.h>
#include <math.h>

// ---------------------------------------------------------------------------
// CDNA5 (gfx1250) fp32 WMMA GEMM pipeline for the GNN prompt model.
// N=8192 nodes, F=H=256, C=40. wave32, V_WMMA_F32_16X16X4_F32,
// double-buffered A tile via GLOBAL_LOAD_ASYNC_TO_LDS_B128 (ASYNCcnt).
// ---------------------------------------------------------------------------

typedef float v2f __attribute__((ext_vector_type(2)));
typedef float v8f __attribute__((ext_vector_type(8)));
typedef int   v4i_b __attribute__((vector_size(16)));   // matches builtin param type

#define KT   32      // K-tile staged through LDS per iteration
#define LDSS 36      // padded LDS row stride (floats): even (float2) & float4-aligned

#define AS1 __attribute__((address_space(1)))
#define AS3 __attribute__((address_space(3)))

#if defined(__HIP_DEVICE_COMPILE__)
#  if __has_builtin(__builtin_amdgcn_global_load_async_to_lds_b128)
#    define ASYNC_COPY 1
#  else
#    define ASYNC_COPY 0
#  endif
#  if __has_builtin(__builtin_amdgcn_s_wait_asynccnt)
#    define WAIT_ASYNC() __builtin_amdgcn_s_wait_asynccnt(0)
#  else
#    define WAIT_ASYNC() asm volatile("s_wait_asynccnt 0" ::: "memory")
#  endif
#else
#  define ASYNC_COPY 0
#  define WAIT_ASYNC()
#endif

enum { EPI_STORE = 0, EPI_BIAS_RELU = 1, EPI_BIAS = 2, EPI_LEAKY = 3, EPI_LEAKY1 = 4 };

// C[64 x 64*NSUB block] = A[64 x K] * B[K x Nout]  (+ epilogue)
// 256 threads = 8 wave32s; each wave owns 32 rows x 16 cols per column subtile.
// A tile: double-buffered in LDS, filled by async-to-LDS copies (unless ZD).
// B tile: single LDS buffer, prefetched to VGPRs before compute, committed after.
template <int EPI, int NSUB, bool ZD>
__global__ __launch_bounds__(256)
void gemm_wmma_f32(const float* __restrict__ A,
                   const float* __restrict__ B,
                   const float* __restrict__ bias,
                   float* __restrict__ C,
                   int K, int Nout, int lda, int ldb, int ldc)
{
    constexpr int W    = 64 * NSUB;            // block tile width
    constexpr int BREP = (KT * W) / (4 * 256); // float4 loads per thread for B

    __shared__ float At[2][64][LDSS];          // A tile x2, row-major (row, k)
    __shared__ float Bt[W][LDSS];              // B tile, transposed (col, k)

    const int tid  = threadIdx.x;
    const int lane = tid & 31;
    const int wid  = tid >> 5;
    const int half = lane >> 4;   // lanes 0-15 vs 16-31
    const int lr   = lane & 15;
    const int wr   = wid & 1;     // wave row within block (0..1 -> 32 rows each)
    const int wc   = wid >> 1;    // wave col within 64-wide subtile (0..3)
    const int row0 = blockIdx.x * 64;
    const int col0 = blockIdx.y * W;

    const v8f vzero = {0.f, 0.f, 0.f, 0.f, 0.f, 0.f, 0.f, 0.f};
    v8f acc[NSUB][2];
#pragma unroll
    for (int s = 0; s < NSUB; ++s) { acc[s][0] = vzero; acc[s][1] = vzero; }

    float4 aReg[2];
    float4 bReg[BREP];

    // ---- issue A tile fetch: async straight into LDS, or VGPR stage (ZD) ----
    auto stage_A = [&](int k0, int buf) {
#pragma unroll
        for (int rep = 0; rep < 2; ++rep) {
            int lin = (tid + rep * 256) * 4;
            int r   = lin >> 5;           // / KT
            int kk  = lin & (KT - 1);
            const float* gp = A + (size_t)(row0 + r) * lda + (k0 + kk);
#if ASYNC_COPY
            if (!ZD) {
                __builtin_amdgcn_global_load_async_to_lds_b128(
                    (AS1 v4i_b*)gp,
                    (AS3 v4i_b*)&At[buf][r][kk], 0, 0);
                continue;
            }
#endif
            float4 v = *(const float4*)gp;
            if (ZD) {                     // adj_a.fill_diagonal_(0)
                int d = (row0 + r) - (k0 + kk);
                if (d == 0) v.x = 0.f;
                if (d == 1) v.y = 0.f;
                if (d == 2) v.z = 0.f;
                if (d == 3) v.w = 0.f;
            }
            aReg[rep] = v;
        }
    };
    auto commit_A = [&](int buf) {        // only needed for the VGPR-staged path
#if ASYNC_COPY
        if (!ZD) return;
#endif
#pragma unroll
        for (int rep = 0; rep < 2; ++rep) {
            int lin = (tid + rep * 256) * 4;
            int r   = lin >> 5;
            int kk  = lin & (KT - 1);
            *(float4*)&At[buf][r][kk] = aReg[rep];
        }
    };
    auto wait_async = [&]() {
#if ASYNC_COPY
        if (!ZD) WAIT_ASYNC();
#endif
    };

    // ---- B tile: prefetch to regs / commit transposed into LDS ----
    auto load_B = [&](int k0) {
#pragma unroll
        for (int rep = 0; rep < BREP; ++rep) {
            int lin = (tid + rep * 256) * 4;
            int kr  = lin / W;
            int n   = lin % W;
            int gk  = k0 + kr;
            float4 v;
            if (col0 + n + 3 < Nout) {
                v = *(const float4*)(B + (size_t)gk * ldb + (col0 + n));
            } else {
                v.x = (col0 + n + 0 < Nout) ? B[(size_t)gk * ldb + col0 + n + 0] : 0.f;
                v.y = (col0 + n + 1 < Nout) ? B[(size_t)gk * ldb + col0 + n + 1] : 0.f;
                v.z = (col0 + n + 2 < Nout) ? B[(size_t)gk * ldb + col0 + n + 2] : 0.f;
                v.w = (col0 + n + 3 < Nout) ? B[(size_t)gk * ldb + col0 + n + 3] : 0.f;
            }
            bReg[rep] = v;
        }
    };
    auto commit_B = [&]() {
#pragma unroll
        for (int rep = 0; rep < BREP; ++rep) {
            int lin = (tid + rep * 256) * 4;
            int kr  = lin / W;
            int n   = lin % W;
            Bt[n + 0][kr] = bReg[rep].x;
            Bt[n + 1][kr] = bReg[rep].y;
            Bt[n + 2][kr] = bReg[rep].z;
            Bt[n + 3][kr] = bReg[rep].w;
        }
    };

    // ---- WMMA compute over one staged K-tile ----
    auto compute = [&](int buf) {
#pragma unroll
        for (int kb = 0; kb < KT; kb += 4) {
            // A 16x4 frag: a[v] = A[row=lr][k = kb + 2*half + v]
            v2f a0 = *(const v2f*)&At[buf][wr * 32 +      lr][kb + half * 2];
            v2f a1 = *(const v2f*)&At[buf][wr * 32 + 16 + lr][kb + half * 2];
#pragma unroll
            for (int s = 0; s < NSUB; ++s) {
                // B 4x16 frag: b[v] = B[k = kb + 2*half + v][col=lr]
                v2f b = *(const v2f*)&Bt[s * 64 + wc * 16 + lr][kb + half * 2];
                acc[s][0] = __builtin_amdgcn_wmma_f32_16x16x4_f32(
                    false, a0, false, b, (short)0, acc[s][0], false, false);
                acc[s][1] = __builtin_amdgcn_wmma_f32_16x16x4_f32(
                    false, a1, false, b, (short)0, acc[s][1], false, false);
            }
        }
    };

    // ---- pipelined main loop ----
    stage_A(0, 0);
    load_B(0);
    commit_A(0);
    commit_B();
    wait_async();
    __syncthreads();

    const int nIter = K / KT;
    for (int i = 0; i < nIter; ++i) {
        const int  buf  = i & 1;
        const bool more = (i + 1) < nIter;
        if (more) {
            stage_A((i + 1) * KT, buf ^ 1);   // async copy overlaps compute
            load_B((i + 1) * KT);             // global latency hidden under WMMAs
        }
        compute(buf);
        __syncthreads();                      // all waves done reading Bt
        if (more) { commit_A(buf ^ 1); commit_B(); }
        wait_async();                         // next A tile landed in LDS
        __syncthreads();                      // new Bt / At[buf^1] visible
    }

    // ---- epilogue: C/D layout row = 8*half + i, col = lr ----
#pragma unroll
    for (int s = 0; s < NSUB; ++s) {
        int gc = col0 + s * 64 + wc * 16 + lr;
        if (gc >= Nout) continue;
        float bv = (EPI == EPI_BIAS_RELU || EPI == EPI_BIAS) ? bias[gc] : 0.f;
#pragma unroll
        for (int t = 0; t < 2; ++t) {
#pragma unroll
            for (int i = 0; i < 8; ++i) {
                int gr  = row0 + wr * 32 + t * 16 + half * 8 + i;
                float v = acc[s][t][i];
                if (EPI == EPI_BIAS_RELU)   v = fmaxf(v + bv, 0.f);
                else if (EPI == EPI_BIAS)   v = v + bv;
                else if (EPI == EPI_LEAKY)  v = (v > 0.f) ? v : 0.2f * v;
                else if (EPI == EPI_LEAKY1) v = ((v > 0.f) ? v : 0.2f * v) + 1.f;
                C[(size_t)gr * ldc + gc] = v;
            }
        }
    }
}

// ---------------------------------------------------------------------------
// lp = gamma*base + beta; lp /= max(||lp||_2, 1e-12); x_in = 0.5*(x + lp)
// one 256-thread block per row (F=256).
// ---------------------------------------------------------------------------
__global__ __launch_bounds__(256)
void fuse_prompt(const float* __restrict__ x, const float* __restrict__ base,
                 const float* __restrict__ gamma, const float* __restrict__ beta,
                 float* __restrict__ x_in)
{
    const int r = blockIdx.x;
    const int f = threadIdx.x;
    const size_t idx = (size_t)r * 256 + f;

    float lp = gamma[idx] * base[idx] + beta[idx];
    float s  = lp * lp;
#pragma unroll
    for (int off = 16; off > 0; off >>= 1) s += __shfl_xor(s, off, 32);

    __shared__ float partial[8];
    if ((threadIdx.x & 31) == 0) partial[threadIdx.x >> 5] = s;
    __syncthreads();
    float tot = partial[0] + partial[1] + partial[2] + partial[3] +
                partial[4] + partial[5] + partial[6] + partial[7];
    float rinv = 1.0f / fmaxf(sqrtf(tot), 1e-12f);
    x_in[idx] = 0.5f * (x[idx] + lp * rinv);
}

// ---------------------------------------------------------------------------
// in-place row log_softmax over C=40 columns; one wave32 per row.
// ---------------------------------------------------------------------------
__global__ __launch_bounds__(256)
void logsoftmax_rows(float* __restrict__ out, int C)
{
    const int wid  = threadIdx.x >> 5;
    const int lane = threadIdx.x & 31;
    const int r    = blockIdx.x * 8 + wid;

    const size_t rowoff = (size_t)r * C;
    float v0 = (lane      < C) ? out[rowoff + lane]      : -INFINITY;
    float v1 = (lane + 32 < C) ? out[rowoff + lane + 32] : -INFINITY;

    float m = fmaxf(v0, v1);
#pragma unroll
    for (int off = 16; off > 0; off >>= 1) m = fmaxf(m, __shfl_xor(m, off, 32));

    float e = ((lane      < C) ? expf(v0 - m) : 0.f) +
              ((lane + 32 < C) ? expf(v1 - m) : 0.f);
#pragma unroll
    for (int off = 16; off > 0; off >>= 1) e += __shfl_xor(e, off, 32);

    float lse = m + logf(e);
    if (lane      < C) out[rowoff + lane]      = v0 - lse;
    if (lane + 32 < C) out[rowoff + lane + 32] = v1 - lse;
}

// ---------------------------------------------------------------------------
extern "C" void kernel_launch(void* const* d_in, const int* in_sizes, int n_in,
                              void* d_out, int out_size, void* d_ws, size_t ws_size,
                              hipStream_t stream)
{
    const float* x    = (const float*)d_in[0];
    const float* adjf = (const float*)d_in[1];
    const float* adja = (const float*)d_in[2];
    const float* Wp   = (const float*)d_in[3];
    const float* bp   = (const float*)d_in[4];
    const float* Wg   = (const float*)d_in[5];
    const float* Wb   = (const float*)d_in[6];
    const float* W1   = (const float*)d_in[7];
    const float* b1   = (const float*)d_in[8];
    const float* W2   = (const float*)d_in[9];
    const float* b2   = (const float*)d_in[10];
    float* out = (float*)d_out;
    float* ws  = (float*)d_ws;

    const int N = 8192, F = 256, C = 40;
    const size_t NF = (size_t)N * F;

    // scratch layout (with reuse): 6 * 8 MB = 48 MB
    float* xp    = ws + 0 * NF;   // x @ Wp            (reused later as t2)
    float* base  = ws + 1 * NF;   // base_prompt
    float* nb    = ws + 2 * NF;   // neighbor          (reused later as h)
    float* gamma = ws + 3 * NF;   // gamma             (reused later as t3)
    float* beta  = ws + 4 * NF;   // beta
    float* xin   = ws + 5 * NF;   // 0.5*(x + learned_prompt)
    float* t2 = xp;               // x_in @ W1
    float* h  = nb;               // relu(adj_a @ t2 + b1)
    float* t3 = gamma;            // h @ W2

    dim3 blk(256);
    dim3 gF(N / 64, 1);           // NSUB=4 -> 256-wide block tile covers all F
    dim3 gC(N / 64, 1);           // NSUB=1 -> 64-wide tile covers C=40

    // 1) xp = x @ Wp
    gemm_wmma_f32<EPI_STORE, 4, false><<<gF, blk, 0, stream>>>(x, Wp, nullptr, xp, F, F, F, F, F);
    // 2) base = relu(adj_f @ xp + bp)
    gemm_wmma_f32<EPI_BIAS_RELU, 4, false><<<gF, blk, 0, stream>>>(adjf, xp, bp, base, N, F, N, F, F);
    // 3) nb = (adj_a with zero diag) @ x
    gemm_wmma_f32<EPI_STORE, 4, true><<<gF, blk, 0, stream>>>(adja, x, nullptr, nb, N, F, N, F, F);
    // 4) gamma = leaky(nb @ Wg) + 1
    gemm_wmma_f32<EPI_LEAKY1, 4, false><<<gF, blk, 0, stream>>>(nb, Wg, nullptr, gamma, F, F, F, F, F);
    // 5) beta = leaky(nb @ Wb)
    gemm_wmma_f32<EPI_LEAKY, 4, false><<<gF, blk, 0, stream>>>(nb, Wb, nullptr, beta, F, F, F, F, F);
    // 6) x_in = 0.5*(x + normalize(gamma*base + beta))
    fuse_prompt<<<dim3(N), blk, 0, stream>>>(x, base, gamma, beta, xin);
    // 7) t2 = x_in @ W1
    gemm_wmma_f32<EPI_STORE, 4, false><<<gF, blk, 0, stream>>>(xin, W1, nullptr, t2, F, F, F, F, F);
    // 8) h = relu(adj_a @ t2 + b1)
    gemm_wmma_f32<EPI_BIAS_RELU, 4, false><<<gF, blk, 0, stream>>>(adja, t2, b1, h, N, F, N, F, F);
    // 9) t3 = h @ W2   (Nout = 40)
    gemm_wmma_f32<EPI_STORE, 1, false><<<gC, blk, 0, stream>>>(h, W2, nullptr, t3, F, C, F, C, C);
    // 10) logits = adj_a @ t3 + b2  -> written straight into d_out
    gemm_wmma_f32<EPI_BIAS, 1, false><<<gC, blk, 0, stream>>>(adja, t3, b2, out, N, C, N, C, C);
    // 11) in-place log_softmax over rows
    logsoftmax_rows<<<dim3(N / 8), blk, 0, stream>>>(out, C);
}